// FCEloss_77300821394028
// MI455X (gfx1250) — compile-verified
//
#include <hip/hip_runtime.h>
#include <cstdint>
#include <cstddef>

// FCE-style distillation loss, MI455X (gfx1250).
// HBM-bound: 262 MB single read of pred -> ~11 us floor @ 23.3 TB/s.
// Strategy: per-wave async staging of each 1000-float row into LDS
// (global_load_async_to_lds_b128, ASYNCcnt double-buffer), then compute
// entirely from LDS/registers in wave32 layout. 1 exp + 0.25 log per
// element keeps us far from the transcendental ceiling.
// Loads carry th:TH_LOAD_NT: the 262 MB stream has zero reuse and exceeds
// the 192 MB L2, so non-temporal policy avoids churning the cache.

#define NCLS 1000
#define WAVES_PER_BLOCK 8
#define ROWS_PER_WAVE 2
#define ROW_PAD_FLOATS 1024   // 4 KB LDS slot per row

__device__ __forceinline__ float wave_max(float v) {
#pragma unroll
  for (int o = 16; o > 0; o >>= 1) v = fmaxf(v, __shfl_xor(v, o, 32));
  return v;
}
__device__ __forceinline__ float wave_sum(float v) {
#pragma unroll
  for (int o = 16; o > 0; o >>= 1) v += __shfl_xor(v, o, 32);
  return v;
}
__device__ __forceinline__ float wave_prod(float v) {
#pragma unroll
  for (int o = 16; o > 0; o >>= 1) v *= __shfl_xor(v, o, 32);
  return v;
}

// One 512-byte coalesced async chunk: 32 lanes x 16 B, global -> LDS.
// CDNA5 ISA: INST_OFFSET is added to BOTH the LDS and global
// addresses, so one immediate advances both sides. Non-temporal hint:
// streamed once, never re-read.
template <int OFS>
__device__ __forceinline__ void async_chunk(uint32_t lds, const char* ga) {
  asm volatile("global_load_async_to_lds_b128 %0, %1, off offset:%2 th:TH_LOAD_NT"
               :: "v"(lds), "v"(ga), "i"(OFS)
               : "memory");
}

// Stage one row (1000 floats = 4000 B) into LDS. 8 instructions; the last
// covers bytes 3584..3999 with lanes 0..25 active (EXEC-masked), so exactly
// 8 ASYNCcnt increments per row.
__device__ __forceinline__ void async_row_to_lds(float* lbuf, const float* gsrc,
                                                 int lane) {
  // addrspace(3) flat pointers carry the LDS byte offset in the low 32 bits.
  uint32_t lds = (uint32_t)(uintptr_t)lbuf + (uint32_t)(lane * 16);
  const char* ga = (const char*)gsrc + (size_t)(lane * 16);
  async_chunk<0>(lds, ga);
  async_chunk<512>(lds, ga);
  async_chunk<1024>(lds, ga);
  async_chunk<1536>(lds, ga);
  async_chunk<2048>(lds, ga);
  async_chunk<2560>(lds, ga);
  async_chunk<3072>(lds, ga);
  if (lane < 26) async_chunk<3584>(lds, ga);
}

__device__ __forceinline__ float row_loss(const float4* buf4, int lane,
                                          int tcls, float wt) {
  // lane L owns classes c = 4*(L + 32k) + m, k=0..7, m=0..3 (c<1000 valid).
  float4 v[8];
#pragma unroll
  for (int k = 0; k < 8; ++k) v[k] = buf4[lane + 32 * k];

  // Pass 1: row max (mask stale tail with select, never arithmetic).
  float mx = -3.402823466e38f;
#pragma unroll
  for (int k = 0; k < 8; ++k) {
    if (k < 7 || lane < 26) {
      mx = fmaxf(mx, fmaxf(fmaxf(v[k].x, v[k].y), fmaxf(v[k].z, v[k].w)));
    }
  }
  mx = wave_max(mx);

  // Pass 2: e_c = exp((pred_c - mx)/TEMP) via exp2; accumulate S.
  const float K = 0.36067376022224085f;  // log2(e)/TEMP, TEMP=4
  float4 e[8];
  float s = 0.f;
#pragma unroll
  for (int k = 0; k < 8; ++k) {
    const bool valid = (k < 7) || (lane < 26);
    float4 x = v[k];
    float4 t;
    t.x = valid ? __builtin_amdgcn_exp2f((x.x - mx) * K) : 0.f;
    t.y = valid ? __builtin_amdgcn_exp2f((x.y - mx) * K) : 0.f;
    t.z = valid ? __builtin_amdgcn_exp2f((x.z - mx) * K) : 0.f;
    t.w = valid ? __builtin_amdgcn_exp2f((x.w - mx) * K) : 0.f;
    e[k] = t;
    s += (t.x + t.y) + (t.z + t.w);
  }
  s = wave_sum(s);
  const float rS = __builtin_amdgcn_rcpf(s);

  // Pass 3: sum log(1-p+eps), 4 factors per log; capture class-t terms.
  // Only one p_c per row can be near 1, so a 4-factor product cannot
  // underflow; grouping cuts log count 4x (250 logs/row).
  const float EPSF = 1e-7f;
  float tot2 = 0.f;   // sum of log2(products)
  float ut = 1.0f;    // (1 - p_t + eps), via product-reduce
  float et = 0.0f;    // e_t, via sum-reduce
#pragma unroll
  for (int k = 0; k < 8; ++k) {
    const bool valid = (k < 7) || (lane < 26);
    const int c0 = 4 * (lane + 32 * k);
    float4 t = e[k];
    float ux = valid ? (1.0f - t.x * rS + EPSF) : 1.0f;
    float uy = valid ? (1.0f - t.y * rS + EPSF) : 1.0f;
    float uz = valid ? (1.0f - t.z * rS + EPSF) : 1.0f;
    float uw = valid ? (1.0f - t.w * rS + EPSF) : 1.0f;
    bool b0 = (c0 + 0 == tcls), b1 = (c0 + 1 == tcls);
    bool b2 = (c0 + 2 == tcls), b3 = (c0 + 3 == tcls);
    ut = b0 ? ux : ut;  et = b0 ? t.x : et;
    ut = b1 ? uy : ut;  et = b1 ? t.y : et;
    ut = b2 ? uz : ut;  et = b2 ? t.z : et;
    ut = b3 ? uw : ut;  et = b3 ? t.w : et;
    tot2 += __builtin_amdgcn_logf((ux * uy) * (uz * uw));
  }
  tot2 = wave_sum(tot2);
  ut = wave_prod(ut);
  et = wave_sum(et);

  const float LN2 = 0.6931471805599453f;
  const float termt = LN2 * __builtin_amdgcn_logf(ut);           // log(1-p_t+eps)
  const float lpt = LN2 * __builtin_amdgcn_logf(et * rS + EPSF); // log(p_t+eps)
  return -(lpt + wt * (LN2 * tot2 - termt));
}

__global__ __launch_bounds__(WAVES_PER_BLOCK * 32) void
fce_kernel(const float* __restrict__ pred, const float* __restrict__ weight,
           const int* __restrict__ teacher, float* __restrict__ out,
           int nrows) {
  __shared__ float sh[WAVES_PER_BLOCK * ROWS_PER_WAVE * ROW_PAD_FLOATS]; // 64 KB
  const int lane = threadIdx.x & 31;
  const int wv = threadIdx.x >> 5;
  const int row0 = (blockIdx.x * WAVES_PER_BLOCK + wv) * ROWS_PER_WAVE;
  if (row0 >= nrows) return;  // nrows is a multiple of ROWS_PER_WAVE

  float* b0 = &sh[(wv * ROWS_PER_WAVE + 0) * ROW_PAD_FLOATS];
  float* b1 = &sh[(wv * ROWS_PER_WAVE + 1) * ROW_PAD_FLOATS];

  // Kick off both rows (16 async instructions outstanding).
  async_row_to_lds(b0, pred + (size_t)row0 * NCLS, lane);
  async_row_to_lds(b1, pred + (size_t)(row0 + 1) * NCLS, lane);

  const int t0 = teacher[row0];
  const int t1 = teacher[row0 + 1];
  const float w0 = weight[t0];
  const float w1 = weight[t1];

  // Async loads complete in order: <=8 outstanding means row0 is in LDS,
  // while row1's 8 loads still overlap with row0's compute.
  asm volatile("s_wait_asynccnt 8" ::: "memory");
  const float l0 = row_loss((const float4*)b0, lane, t0, w0);

  asm volatile("s_wait_asynccnt 0" ::: "memory");
  const float l1 = row_loss((const float4*)b1, lane, t1, w1);

  if (lane == 0) {
    out[row0] = l0;
    out[row0 + 1] = l1;
  }
}

extern "C" void kernel_launch(void* const* d_in, const int* in_sizes, int n_in,
                              void* d_out, int out_size, void* d_ws,
                              size_t ws_size, hipStream_t stream) {
  const float* pred = (const float*)d_in[0];
  const float* weight = (const float*)d_in[1];
  const int* teacher = (const int*)d_in[2];
  float* out = (float*)d_out;

  const int nrows = out_size;  // B = 65536
  const int rows_per_block = WAVES_PER_BLOCK * ROWS_PER_WAVE;  // 16
  const int blocks = (nrows + rows_per_block - 1) / rows_per_block;  // 4096
  fce_kernel<<<blocks, WAVES_PER_BLOCK * 32, 0, stream>>>(pred, weight, teacher,
                                                          out, nrows);
}